// UNet_86406152061381
// MI455X (gfx1250) — compile-verified
//
#include <hip/hip_runtime.h>

typedef __attribute__((ext_vector_type(2))) float v2f;
typedef __attribute__((ext_vector_type(8))) float v8f;

namespace {
constexpr int B_ = 4, C_ = 5, C8 = 8, W_ = 256;
constexpr int HW = 256 * 256;          // 65536
constexpr int N_ = 2048;
constexpr int CAP = 2048;              // padded query-row capacity per batch
constexpr int RTOT = B_ * CAP;         // 8192
constexpr int JCH = 32;                // pixel chunks per row
constexpr int CHPIX = HW / JCH;        // 2048 pixels per chunk

// workspace layout (float offsets)
constexpr size_t KP_OFF   = 0;                                    // B*HW*8 = 2,097,152
constexpr size_t VP_OFF   = KP_OFF + (size_t)B_ * HW * C8;        // 2,097,152
constexpr size_t QP_OFF   = VP_OFF + (size_t)B_ * HW * C8;        // 65,536
constexpr size_t PART_OFF = QP_OFF + (size_t)RTOT * C8;           // 2,097,152
constexpr size_t INT_OFF  = PART_OFF + (size_t)RTOT * JCH * 8;    // ints after this
}

// ---------------------------------------------------------------------------
// Kernel 0: stable-partition queries by batch into padded 16-row groups,
// gather x at sampled pixels, apply Wq conv -> padded Q rows (8 channels).
// ---------------------------------------------------------------------------
__global__ void prep_kernel(const float* __restrict__ x,
                            const float* __restrict__ Wq,
                            const float* __restrict__ bq,
                            const int* __restrict__ idx_b,
                            const int* __restrict__ idx_h,
                            const int* __restrict__ idx_w,
                            float* __restrict__ ws,
                            int* __restrict__ rowmap,
                            int* __restrict__ meta) {
  if (threadIdx.x == 0) {
    int cnt[B_] = {0, 0, 0, 0};
    for (int r = 0; r < RTOT; ++r) rowmap[r] = -1;
    for (int n = 0; n < N_; ++n) {
      int b = idx_b[n];
      rowmap[b * CAP + cnt[b]] = n;
      cnt[b]++;
    }
    for (int b = 0; b < B_; ++b) meta[b] = (cnt[b] + 15) >> 4;  // #16-row groups
  }
  __threadfence_block();
  __syncthreads();

  float* Qp = ws + QP_OFF;
  for (int r = threadIdx.x; r < RTOT; r += blockDim.x) {
    int n = rowmap[r];
    float q[C8];
#pragma unroll
    for (int o = 0; o < C8; ++o) q[o] = 0.0f;
    if (n >= 0) {
      int b = idx_b[n];
      int p = idx_h[n] * W_ + idx_w[n];
      float xv[C_];
#pragma unroll
      for (int c = 0; c < C_; ++c) xv[c] = x[((size_t)(b * C_ + c)) * HW + p];
#pragma unroll
      for (int o = 0; o < C_; ++o) {
        float acc = bq[o];
#pragma unroll
        for (int c = 0; c < C_; ++c) acc += Wq[o * C_ + c] * xv[c];
        q[o] = acc;
      }
    }
#pragma unroll
    for (int o = 0; o < C8; ++o) Qp[(size_t)r * C8 + o] = q[o];
  }
}

// ---------------------------------------------------------------------------
// Kernel 1: K/V 1x1 convs into channel-padded pixel-major layout (8 ch/pixel),
// plus base copy y = x.
// ---------------------------------------------------------------------------
__global__ void kv_kernel(const float* __restrict__ x,
                          const float* __restrict__ Wk,
                          const float* __restrict__ bk,
                          const float* __restrict__ Wv,
                          const float* __restrict__ bv,
                          float* __restrict__ ws,
                          float* __restrict__ y) {
  int gp = blockIdx.x * blockDim.x + threadIdx.x;  // 0 .. B*HW-1
  int b = gp >> 16;
  int p = gp & (HW - 1);

  float xv[C_];
#pragma unroll
  for (int c = 0; c < C_; ++c) xv[c] = x[((size_t)(b * C_ + c)) * HW + p];

  float kk[C8], vv[C8];
#pragma unroll
  for (int o = 0; o < C8; ++o) { kk[o] = 0.0f; vv[o] = 0.0f; }
#pragma unroll
  for (int o = 0; o < C_; ++o) {
    float ak = bk[o], av = bv[o];
#pragma unroll
    for (int c = 0; c < C_; ++c) {
      ak += Wk[o * C_ + c] * xv[c];
      av += Wv[o * C_ + c] * xv[c];
    }
    kk[o] = ak;
    vv[o] = av;
  }

  float* Kp = ws + KP_OFF + (size_t)gp * C8;
  float* Vp = ws + VP_OFF + (size_t)gp * C8;
  *(float4*)(Kp + 0) = make_float4(kk[0], kk[1], kk[2], kk[3]);
  *(float4*)(Kp + 4) = make_float4(kk[4], kk[5], kk[6], kk[7]);
  *(float4*)(Vp + 0) = make_float4(vv[0], vv[1], vv[2], vv[3]);
  *(float4*)(Vp + 4) = make_float4(vv[4], vv[5], vv[6], vv[7]);

#pragma unroll
  for (int c = 0; c < C_; ++c) y[((size_t)(b * C_ + c)) * HW + p] = xv[c];
}

// ---------------------------------------------------------------------------
// Kernel 2: flash-attention partials. Workgroup = 8 waves sharing one
// (batch, 2048-pixel chunk); K and V chunks are staged into LDS once via
// CDNA5 async global->LDS b128 copies (ASYNCcnt), then each wave computes a
// 16-query group: energies via chained V_WMMA_F32_16X16X4_F32 (K padded
// 5->8), two phases (row maxima, then exp/sum/V-accumulate), writing
// (m, s, o[5]) per row per chunk.
// ---------------------------------------------------------------------------
__global__ void __launch_bounds__(256) attn_kernel(const float* __restrict__ ws,
                                                   const int* __restrict__ meta,
                                                   float* __restrict__ part) {
  const int lane = threadIdx.x & 31;
  const int wid = threadIdx.x >> 5;
  const int bb = blockIdx.x >> 4;      // batch (16 group-blocks per batch)
  const int gblk = blockIdx.x & 15;
  const int j = blockIdx.y;

  const int ng = meta[bb];             // block-uniform
  if (gblk * 8 >= ng) return;          // fully-inactive workgroup: exit early
  const int g = gblk * 8 + wid;
  const bool active = (g < ng);        // wave-uniform

  __shared__ __align__(16) float sK[CHPIX * C8];  // 64 KB
  __shared__ __align__(16) float sV[CHPIX * C8];  // 64 KB

  const float* Kp = ws + KP_OFF + (size_t)bb * HW * C8;
  const float* Vp = ws + VP_OFF + (size_t)bb * HW * C8;
  const int p0 = j * CHPIX;

  // ---- cooperative async stage of the contiguous 64KB K and V chunks ----
  {
    const char* gk = (const char*)(Kp + (size_t)p0 * C8);
    const char* gv = (const char*)(Vp + (size_t)p0 * C8);
    const unsigned skb = (unsigned)(uintptr_t)(&sK[0]);
    const unsigned svb = (unsigned)(uintptr_t)(&sV[0]);
#pragma unroll
    for (int it = 0; it < 16; ++it) {
      const unsigned off = ((unsigned)threadIdx.x + (unsigned)it * 256u) * 16u;
      const unsigned long long gka = (unsigned long long)(uintptr_t)(gk + off);
      const unsigned long long gva = (unsigned long long)(uintptr_t)(gv + off);
      asm volatile("global_load_async_to_lds_b128 %0, %1, off"
                   :: "v"(skb + off), "v"(gka) : "memory");
      asm volatile("global_load_async_to_lds_b128 %0, %1, off"
                   :: "v"(svb + off), "v"(gva) : "memory");
    }
    asm volatile("s_wait_asynccnt 0x0" ::: "memory");
  }
  __syncthreads();

  if (!active) return;                 // wave-uniform; no barriers follow

  const int half = lane >> 4;   // 0: rows 0-7 / K 0-1, 1: rows 8-15 / K 2-3
  const int ln = lane & 15;     // column / row-in-half index
  const int rbase = bb * CAP + g * 16;

  const float* Qp = ws + QP_OFF;

  // A operand (16x4 fp32): lanes 0-15 hold K=0,1; lanes 16-31 hold K=2,3.
  const int arow = rbase + ln;
  const int ka = half * 2;
  const v2f a0 = *(const v2f*)(Qp + (size_t)arow * C8 + ka);       // K 0..3
  const v2f a1 = *(const v2f*)(Qp + (size_t)arow * C8 + ka + 4);   // K 4..7

  // -------- phase 1: per-row maxima over this chunk --------
  float rm[8];
#pragma unroll
  for (int i = 0; i < 8; ++i) rm[i] = -1e30f;

  for (int t = 0; t < CHPIX; t += 16) {
    const int px = t + ln;
    const float* kcol = sK + px * C8 + ka;
    const v2f b0 = *(const v2f*)(kcol);       // B rows 0..3 (K dim)
    const v2f b1 = *(const v2f*)(kcol + 4);   // B rows 4..7
    v8f e = {};
    e = __builtin_amdgcn_wmma_f32_16x16x4_f32(false, a0, false, b0, (short)0, e,
                                              false, false);
    e = __builtin_amdgcn_wmma_f32_16x16x4_f32(false, a1, false, b1, (short)0, e,
                                              false, false);
#pragma unroll
    for (int i = 0; i < 8; ++i) rm[i] = fmaxf(rm[i], e[i]);
  }
#pragma unroll
  for (int i = 0; i < 8; ++i) {
#pragma unroll
    for (int off = 1; off < 16; off <<= 1)
      rm[i] = fmaxf(rm[i], __shfl_xor(rm[i], off, 16));
  }

  // -------- phase 2: exp-sum and V accumulation --------
  float s[8];
  float o[8][C_];
#pragma unroll
  for (int i = 0; i < 8; ++i) {
    s[i] = 0.0f;
#pragma unroll
    for (int c = 0; c < C_; ++c) o[i][c] = 0.0f;
  }

  for (int t = 0; t < CHPIX; t += 16) {
    const int px = t + ln;
    const float* kcol = sK + px * C8 + ka;
    const v2f b0 = *(const v2f*)(kcol);
    const v2f b1 = *(const v2f*)(kcol + 4);
    v8f e = {};
    e = __builtin_amdgcn_wmma_f32_16x16x4_f32(false, a0, false, b0, (short)0, e,
                                              false, false);
    e = __builtin_amdgcn_wmma_f32_16x16x4_f32(false, a1, false, b1, (short)0, e,
                                              false, false);
    const float4 v4 = *(const float4*)(sV + px * C8);
    const float v5 = sV[px * C8 + 4];
#pragma unroll
    for (int i = 0; i < 8; ++i) {
      const float w = __expf(e[i] - rm[i]);
      s[i] += w;
      o[i][0] += w * v4.x;
      o[i][1] += w * v4.y;
      o[i][2] += w * v4.z;
      o[i][3] += w * v4.w;
      o[i][4] += w * v5;
    }
  }

  // reduce across the 16 lanes of each half
#pragma unroll
  for (int i = 0; i < 8; ++i) {
#pragma unroll
    for (int off = 1; off < 16; off <<= 1) {
      s[i] += __shfl_xor(s[i], off, 16);
#pragma unroll
      for (int c = 0; c < C_; ++c) o[i][c] += __shfl_xor(o[i][c], off, 16);
    }
  }

  if (ln == 0) {
#pragma unroll
    for (int i = 0; i < 8; ++i) {
      const int r = rbase + half * 8 + i;
      float* dst = part + ((size_t)r * JCH + j) * 8;
      dst[0] = rm[i];
      dst[1] = s[i];
#pragma unroll
      for (int c = 0; c < C_; ++c) dst[2 + c] = o[i][c];
    }
  }
}

// ---------------------------------------------------------------------------
// Kernel 3: merge flash partials across chunks, scatter updates into y.
// ---------------------------------------------------------------------------
__global__ void combine_kernel(const float* __restrict__ part,
                               const int* __restrict__ rowmap,
                               const float* __restrict__ x,
                               const float* __restrict__ gamma,
                               const int* __restrict__ idx_b,
                               const int* __restrict__ idx_h,
                               const int* __restrict__ idx_w,
                               float* __restrict__ y) {
  const int r = blockIdx.x * blockDim.x + threadIdx.x;
  if (r >= RTOT) return;
  const int n = rowmap[r];
  if (n < 0) return;

  float m = -1e30f;
  for (int j = 0; j < JCH; ++j)
    m = fmaxf(m, part[((size_t)r * JCH + j) * 8]);

  float S = 0.0f;
  float O[C_] = {0.f, 0.f, 0.f, 0.f, 0.f};
  for (int j = 0; j < JCH; ++j) {
    const float* pp = part + ((size_t)r * JCH + j) * 8;
    const float sc = __expf(pp[0] - m);
    S += pp[1] * sc;
#pragma unroll
    for (int c = 0; c < C_; ++c) O[c] += pp[2 + c] * sc;
  }
  const float inv = 1.0f / S;

  const int b = idx_b[n];
  const int p = idx_h[n] * W_ + idx_w[n];
  const float gsc = gamma[0] + 0.1f;
#pragma unroll
  for (int c = 0; c < C_; ++c) {
    const float outc = O[c] * inv;
    const size_t off = ((size_t)(b * C_ + c)) * HW + p;
    y[off] = gsc * outc + x[off];
  }
}

// ---------------------------------------------------------------------------
extern "C" void kernel_launch(void* const* d_in, const int* in_sizes, int n_in,
                              void* d_out, int out_size, void* d_ws,
                              size_t ws_size, hipStream_t stream) {
  const float* x = (const float*)d_in[0];
  const float* Wq = (const float*)d_in[1];
  const float* bq = (const float*)d_in[2];
  const float* Wk = (const float*)d_in[3];
  const float* bk = (const float*)d_in[4];
  const float* Wv = (const float*)d_in[5];
  const float* bv = (const float*)d_in[6];
  const float* gamma = (const float*)d_in[7];
  const int* idx_b = (const int*)d_in[8];
  const int* idx_h = (const int*)d_in[9];
  const int* idx_w = (const int*)d_in[10];

  float* ws = (float*)d_ws;
  float* y = (float*)d_out;
  float* part = ws + PART_OFF;
  int* iws = (int*)(ws + INT_OFF);
  int* rowmap = iws;
  int* meta = iws + RTOT;

  prep_kernel<<<1, 256, 0, stream>>>(x, Wq, bq, idx_b, idx_h, idx_w, ws, rowmap,
                                     meta);
  kv_kernel<<<(B_ * HW) / 256, 256, 0, stream>>>(x, Wk, bk, Wv, bv, ws, y);
  attn_kernel<<<dim3(64, JCH), 256, 0, stream>>>(ws, meta, part);
  combine_kernel<<<RTOT / 256, 256, 0, stream>>>(part, rowmap, x, gamma, idx_b,
                                                 idx_h, idx_w, y);
}